// FastVoxelizer_57062935495124
// MI455X (gfx1250) — compile-verified
//
#include <hip/hip_runtime.h>
#include <hip/hip_bf16.h>

typedef __attribute__((ext_vector_type(2))) float v2f;
typedef __attribute__((ext_vector_type(8))) float v8f;

#define NMAX 512          // max gaussians stageable in LDS (ref: 400)
#define CFEAT 32          // feature channels (ref: 32)
#define WAVES_PER_BLOCK 4 // 4 wave32 => 128 threads, 64 voxels / block

// ---------------------------------------------------------------------------
// Kernel 1: per-gaussian precompute.
// Inverts the symmetric 3x3 covariance via adjugate and pre-folds the
// -0.5 (diagonal) / -1.0 (off-diagonal, absorbs the 2x) factors so the hot
// loop is exp(q.x*dx^2 + ... ) directly.  thresh = (3*sigma_max)^2 = 9*max(diag).
// Packs 12 floats per gaussian (3x float4) into d_ws.
// ---------------------------------------------------------------------------
__global__ void fv_precompute(const float* __restrict__ means,
                              const float* __restrict__ opac,
                              const float* __restrict__ cov,
                              float* __restrict__ gp, int N) {
    int i = blockIdx.x * blockDim.x + threadIdx.x;
    if (i >= N) return;
    const float* Cm = cov + (size_t)i * 9;
    float a = Cm[0], b = Cm[1], c = Cm[2];
    float d = Cm[4], e = Cm[5], f = Cm[8];
    float A0 = d * f - e * e;
    float B0 = c * e - b * f;
    float C0 = b * e - c * d;
    float det = a * A0 + b * B0 + c * C0;
    float r = 1.0f / det;
    // folded inverse: diag * -0.5, off-diag * -1.0 (covers the 2x cross terms)
    float4 q0 = { -0.5f * A0 * r, -B0 * r, -C0 * r, -0.5f * (a * f - c * c) * r };
    float4 q1 = { -(b * c - a * e) * r, -0.5f * (a * d - b * b) * r,
                  means[i * 3 + 0], means[i * 3 + 1] };
    float4 q2 = { means[i * 3 + 2], opac[i], 9.0f * fmaxf(a, fmaxf(d, f)), 0.0f };
    float4* o = (float4*)gp + (size_t)i * 3;
    o[0] = q0; o[1] = q1; o[2] = q2;
}

// ---------------------------------------------------------------------------
// Kernel 2: one wave32 per 16-voxel tile.
//   A (16x4, f32): lane half h supplies K = {2h, 2h+1} as {a.x, a.y}  (ISA A layout)
//   B (4x16, f32): lane half h supplies rows {2h, 2h+1} at column lane&15
//   D (16x16, f32): VGPR r holds M = r + 8*h, N = lane&15
// acc0 = feature cols 0..15, acc1 = cols 16..31.
// ---------------------------------------------------------------------------
__global__ void __launch_bounds__(128)
fv_main(const float* __restrict__ gp,     // N*12 packed params
        const float* __restrict__ feats,  // N x 32
        const float* __restrict__ grid,   // V x 3
        float* __restrict__ out,          // [0,V): density  [V, V+V*32): feats
        int N, int V) {
    __shared__ float4 sP[NMAX * 3];                // 24.6 KB
    __shared__ float  sDens[WAVES_PER_BLOCK * 16]; // 256 B

    // stage per-gaussian params into LDS (cooperative, whole block)
    const float4* gp4 = (const float4*)gp;
    int nq = N * 3;
    for (int t = threadIdx.x; t < nq; t += blockDim.x) sP[t] = gp4[t];
    __syncthreads();

    const int lane = threadIdx.x & 31;
    const int wave = threadIdx.x >> 5;
    const int m    = lane & 15;   // voxel row (A/M) and feature col (B/N)
    const int half = lane >> 4;   // K-pair selector
    const int base = (blockIdx.x * WAVES_PER_BLOCK + wave) * 16;

    int v = base + m; if (v > V - 1) v = V - 1;   // clamp; stores are guarded
    const float gx = grid[v * 3 + 0];
    const float gy = grid[v * 3 + 1];
    const float gz = grid[v * 3 + 2];

    v8f acc0 = {}; v8f acc1 = {};
    float dsum = 0.0f;

    const int Npad = (N + 3) & ~3;
    for (int k0 = 0; k0 < Npad; k0 += 4) {
        const int g0 = k0 + half * 2;
        v2f a, b0, b1;
#pragma unroll
        for (int j = 0; j < 2; ++j) {
            const int g  = g0 + j;
            const int gc = (g < N - 1) ? g : (N - 1);   // clamped index for loads
            const float4 q0 = sP[gc * 3 + 0];
            const float4 q1 = sP[gc * 3 + 1];
            const float4 q2 = sP[gc * 3 + 2];
            const float dx = q1.z - gx, dy = q1.w - gy, dz = q2.x - gz;
            const float mh = q0.x * dx * dx + q0.w * dy * dy + q1.y * dz * dz
                           + q0.y * dx * dy + q0.z * dx * dz + q1.x * dy * dz;
            const float dsq = dx * dx + dy * dy + dz * dz;
            float cv = q2.y * __expf(mh);              // v_exp_f32
            cv = (dsq < q2.z) ? cv : 0.0f;             // sigma cutoff
            cv = (g < N) ? cv : 0.0f;                  // K padding
            const float fx0 = feats[gc * CFEAT + m];
            const float fx1 = feats[gc * CFEAT + 16 + m];
            if (j == 0) { a.x = cv; b0.x = fx0; b1.x = fx1; }
            else        { a.y = cv; b0.y = fx0; b1.y = fx1; }
            dsum += cv;
        }
        // D = A x B + C, full fp32 (EXEC is all-ones: no divergent flow here)
        acc0 = __builtin_amdgcn_wmma_f32_16x16x4_f32(false, a, false, b0,
                                                     (short)0, acc0, false, false);
        acc1 = __builtin_amdgcn_wmma_f32_16x16x4_f32(false, a, false, b1,
                                                     (short)0, acc1, false, false);
    }

    // density: combine the two half-wave K partitions (wave32 shuffle)
    const float dtot = dsum + __shfl_xor(dsum, 16, 32);
    if (lane < 16) {
        sDens[wave * 16 + m] = dtot;
        if (base + m < V) out[base + m] = dtot;
    }
    __syncthreads();

    // normalize + store feats: VGPR r of C/D holds voxel row r + 8*half
#pragma unroll
    for (int rI = 0; rI < 8; ++rI) {
        const int mm = rI + half * 8;
        const int vv = base + mm;
        if (vv < V) {
            const float dd = sDens[wave * 16 + mm];
            const float s  = 1.0f / fmaxf(dd, 1e-6f);
            out[(size_t)V + (size_t)vv * CFEAT + m]      = acc0[rI] * s;
            out[(size_t)V + (size_t)vv * CFEAT + 16 + m] = acc1[rI] * s;
        }
    }
}

// ---------------------------------------------------------------------------
extern "C" void kernel_launch(void* const* d_in, const int* in_sizes, int n_in,
                              void* d_out, int out_size, void* d_ws, size_t ws_size,
                              hipStream_t stream) {
    const float* means = (const float*)d_in[0];   // N x 3
    const float* opac  = (const float*)d_in[1];   // N x 1
    const float* cov   = (const float*)d_in[2];   // N x 3 x 3
    const float* feat  = (const float*)d_in[3];   // N x 32
    const float* grid  = (const float*)d_in[4];   // V x 3
    float* out = (float*)d_out;

    const int N = in_sizes[0] / 3;
    const int V = in_sizes[4] / 3;
    float* gp = (float*)d_ws;                     // N*12 floats of scratch

    fv_precompute<<<(N + 127) / 128, 128, 0, stream>>>(means, opac, cov, gp, N);

    const int numTiles = (V + 15) / 16;
    const int blocks   = (numTiles + WAVES_PER_BLOCK - 1) / WAVES_PER_BLOCK;
    fv_main<<<blocks, 128, 0, stream>>>(gp, feat, grid, out, N, V);
}